// VolatilityTrackingLayer_50483045598000
// MI455X (gfx1250) — compile-verified
//
#include <hip/hip_runtime.h>
#include <hip/hip_bf16.h>

typedef __attribute__((ext_vector_type(2))) float v2f;
typedef __attribute__((ext_vector_type(8))) float v8f;

#define LN2F 0.6931471805599453f
#define EPSF 1e-08f

// x: (B, 4, 256) f32 channels [open, high, low, close]; window = last 64 steps.
// out: (B, 256) f32 = relu(vm @ W^T + b), vm = [std_dev, gk, pk] per row.
__global__ __launch_bounds__(256)
void vol_track_kernel(const float* __restrict__ x,
                      const float* __restrict__ Wlin,   // (256, 3)
                      const float* __restrict__ blin,   // (256,)
                      float* __restrict__ out)          // (B, 256)
{
    const int lane = threadIdx.x & 31;
    const int wave = threadIdx.x >> 5;
    const int row0 = blockIdx.x * 128 + wave * 16;   // 16 batch rows per wave
    const bool lo16 = (lane < 16);

    // ---- stage 1: per-row volatility reductions, 2 lanes per row ----
    const int r    = lane >> 1;        // row within tile, 0..15
    const int half = lane & 1;         // which 32-step chunk of the 64 window
    const long rowbase = (long)(row0 + r) * 1024;    // 4*256 floats per batch row

    const float* __restrict__ highp  = x + rowbase + 1 * 256 + 192 + 32 * half;
    const float* __restrict__ lowp   = x + rowbase + 2 * 256 + 192 + 32 * half;
    const float* __restrict__ closep = x + rowbase + 3 * 256 + 192 + 32 * half;

    float S1 = 0.0f, S2 = 0.0f, S3 = 0.0f;   // sum logCC, sum logCC^2, sum logHL^2

    // log(high/low + eps)^2 over this lane's 32 steps (float4 = global_load_b128)
    const float4* __restrict__ h4 = (const float4*)highp;
    const float4* __restrict__ l4 = (const float4*)lowp;
    #pragma unroll
    for (int k = 0; k < 8; ++k) {
        float4 hv = h4[k], lv = l4[k];
        float a0 = __logf(hv.x / lv.x + EPSF);
        float a1 = __logf(hv.y / lv.y + EPSF);
        float a2 = __logf(hv.z / lv.z + EPSF);
        float a3 = __logf(hv.w / lv.w + EPSF);
        S3 += a0 * a0 + a1 * a1 + a2 * a2 + a3 * a3;
    }

    // close-to-close log returns: 63 ratios over the 64-window.
    // half=0 covers j=0..31 (needs close[32] extra), half=1 covers j=32..62.
    float c[33];
    const float4* __restrict__ c4 = (const float4*)closep;
    #pragma unroll
    for (int k = 0; k < 8; ++k) {
        float4 cv = c4[k];
        c[4 * k + 0] = cv.x; c[4 * k + 1] = cv.y;
        c[4 * k + 2] = cv.z; c[4 * k + 3] = cv.w;
    }
    // Branchless extra element: half=0 needs close[32]; half=1 reads close[31]
    // (in-range dummy, its ratio contribution is selected out below).
    c[32] = closep[(half == 0) ? 32 : -1];

    #pragma unroll
    for (int j = 0; j < 32; ++j) {
        float nxt = (j < 31) ? c[j + 1] : c[32];
        float lcc = __logf(nxt / c[j] + EPSF);
        lcc = (j == 31 && half == 1) ? 0.0f : lcc;   // only 63 ratios total
        S1 += lcc;
        S2 += lcc * lcc;
    }

    // combine the two half-lanes of each row
    S1 += __shfl_xor(S1, 1, 32);
    S2 += __shfl_xor(S2, 1, 32);
    S3 += __shfl_xor(S3, 1, 32);

    const float inv63 = 1.0f / 63.0f;
    float var_cc   = (S2 - S1 * S1 * inv63) * (1.0f / 62.0f);
    float std_dev  = sqrtf(fmaxf(var_cc, 0.0f));               // nan_to_num -> 0
    float mean_hl2 = S3 * (1.0f / 64.0f);
    float mean_cc2 = S2 * inv63;
    float gk2      = 0.5f * mean_hl2 - (2.0f * LN2F - 1.0f) * mean_cc2;
    float gk       = sqrtf(fmaxf(gk2, 0.0f));                  // nan_to_num -> 0
    float pk       = sqrtf(mean_hl2 * (1.0f / (4.0f * LN2F)));

    // ---- redistribute into V_WMMA_F32_16X16X4_F32 A-fragment layout ----
    // lanes 0-15:  M=lane,    VGPR0=K0(std), VGPR1=K1(gk)
    // lanes 16-31: M=lane-16, VGPR0=K2(pk),  VGPR1=K3(0 pad)
    const int m   = lane & 15;
    float stdm = __shfl(std_dev, 2 * m, 32);
    float gkm  = __shfl(gk,      2 * m, 32);
    float pkm  = __shfl(pk,      2 * m, 32);

    v2f afrag;
    afrag.x = lo16 ? stdm : pkm;
    afrag.y = lo16 ? gkm  : 0.0f;

    const int ksel = lo16 ? 0 : 2;   // which K element this half-wave holds

    // ---- stage 2: out[16 rows x 256 cols] via 16 WMMA f32 16x16x4 ----
    #pragma unroll
    for (int nt = 0; nt < 16; ++nt) {
        const int n0 = nt * 16;
        const int wrow = (n0 + m) * 3;   // W_lin row for this lane's column

        // B-fragment (4x16, K-major mirror of A): lanes 0-15 hold K0,K1 of
        // column n0+m; lanes 16-31 hold K2,K3(=0). B[k][n] = W[n][k].
        // Branchless: index select for .x, unconditional load + cndmask for .y
        // (wrow+1 <= 766 < 768 is always in range).
        v2f bfrag;
        bfrag.x = Wlin[wrow + ksel];
        float wk1 = Wlin[wrow + 1];
        bfrag.y = lo16 ? wk1 : 0.0f;

        // C = bias broadcast: C/D layout has N = lane&15 in every VGPR.
        float bias = blin[n0 + m];
        v8f acc;
        #pragma unroll
        for (int i = 0; i < 8; ++i) acc[i] = bias;

        v8f d = __builtin_amdgcn_wmma_f32_16x16x4_f32(
            /*neg_a=*/false, afrag, /*neg_b=*/false, bfrag,
            /*c_mod=*/(short)0, acc, /*reuse_a=*/false, /*reuse_b=*/false);

        // D layout: VGPR i -> M = i (+8 for lanes 16-31), N = lane&15
        const int rbase = row0 + (lo16 ? 0 : 8);
        #pragma unroll
        for (int i = 0; i < 8; ++i) {
            out[(long)(rbase + i) * 256 + n0 + m] = fmaxf(d[i], 0.0f);
        }
    }
}

extern "C" void kernel_launch(void* const* d_in, const int* in_sizes, int n_in,
                              void* d_out, int out_size, void* d_ws, size_t ws_size,
                              hipStream_t stream)
{
    const float* x    = (const float*)d_in[0];   // (65536, 4, 256)
    const float* Wlin = (const float*)d_in[1];   // (256, 3)
    const float* blin = (const float*)d_in[2];   // (256,)
    // d_in[3] = window_size == 64 (hardcoded: min(64, T) == 64)
    float* out = (float*)d_out;                  // (65536, 256)

    // 128 rows per block (8 waves x 16 rows) -> 512 blocks
    vol_track_kernel<<<dim3(512), dim3(256), 0, stream>>>(x, Wlin, blin, out);
}